// Pruning_39848706572522
// MI455X (gfx1250) — compile-verified
//
#include <hip/hip_runtime.h>
#include <stdint.h>

#define B        32
#define N        (1u << 20)
#define K        4096
#define NBINS    8192
#define BIN_SHIFT 19            // key >> 19 -> 13-bit bin
#define CAND_CAP 8192
#define SORT_N   8192
#define HIST_BYTES (B * NBINS * 4)   // 1 MB

typedef __attribute__((address_space(3))) unsigned char lds_byte_t;

// Order-preserving float -> uint32 key (larger key == larger float)
__device__ __forceinline__ uint32_t fkey(float f) {
    uint32_t u = __float_as_uint(f);
    return (u & 0x80000000u) ? ~u : (u | 0x80000000u);
}

__device__ __forceinline__ void async_load_b128(uint32_t lds_off, const void* gptr) {
    // CDNA5 async copy: global -> LDS, tracked by ASYNCcnt (per-lane 16B)
    asm volatile("global_load_async_to_lds_b128 %0, %1, off"
                 :: "v"(lds_off), "v"((unsigned long long)(uintptr_t)gptr)
                 : "memory");
}
__device__ __forceinline__ void wait_async0() {
    asm volatile("s_wait_asynccnt 0x0" ::: "memory");
}

// ---------------- Kernel 1: zero histograms + candidate counters ----------------
__global__ void k_init(uint32_t* __restrict__ hist, uint32_t* __restrict__ cnt) {
    uint32_t i = blockIdx.x * blockDim.x + threadIdx.x;
    if (i < B * NBINS) hist[i] = 0u;
    if (i < B)         cnt[i]  = 0u;
}

// ---------------- Kernel 2: per-row 8192-bin histogram (async LDS staging) ------
__global__ void k_hist(const float* __restrict__ scores, uint32_t* __restrict__ hist) {
    __shared__ uint32_t shist[NBINS];                  // 32 KB
    __shared__ __align__(16) float tile[1024];         // 4 KB staging tile
    const int tid = threadIdx.x;                       // 256 threads
    for (int i = tid; i < NBINS; i += 256) shist[i] = 0u;
    __syncthreads();

    const int row = blockIdx.x >> 8;                   // 256 blocks per row
    const int blk = blockIdx.x & 255;                  // 4096 elements per block
    const unsigned long long rowbase = (unsigned long long)row * N;

    const uint32_t tile_lds =
        (uint32_t)(unsigned long long)(lds_byte_t*)tile + (uint32_t)tid * 16u;

    for (int r = 0; r < 4; ++r) {                      // 4 x 1024 floats
        const uint32_t e = (uint32_t)blk * 4096u + (uint32_t)r * 1024u + (uint32_t)tid * 4u;
        const float* gp = scores + rowbase + e;
        async_load_b128(tile_lds, gp);                 // ASYNCcnt++
        wait_async0();                                 // data landed in LDS
        float4 v = *(const float4*)&tile[tid * 4];
        atomicAdd(&shist[fkey(v.x) >> BIN_SHIFT], 1u);
        atomicAdd(&shist[fkey(v.y) >> BIN_SHIFT], 1u);
        atomicAdd(&shist[fkey(v.z) >> BIN_SHIFT], 1u);
        atomicAdd(&shist[fkey(v.w) >> BIN_SHIFT], 1u);
    }
    __syncthreads();

    uint32_t* gh = hist + (size_t)row * NBINS;
    for (int i = tid; i < NBINS; i += 256) {
        uint32_t c = shist[i];
        if (c) atomicAdd(&gh[i], c);
    }
}

// ---------------- Kernel 3: find threshold bin per row --------------------------
__global__ void k_thresh(const uint32_t* __restrict__ hist, uint32_t* __restrict__ thresh) {
    __shared__ uint32_t seg[256];
    const int row = blockIdx.x;
    const int tid = threadIdx.x;                       // 256 threads
    const uint32_t* h = hist + (size_t)row * NBINS;
    uint32_t s = 0;
    for (int i = 0; i < 32; ++i) s += h[tid * 32 + i]; // 32-bin segment sums
    seg[tid] = s;
    __syncthreads();
    if (tid == 0) {
        uint32_t acc = 0, t = 0;
        bool found = false;
        for (int sg = 255; sg >= 0 && !found; --sg) {
            if (acc + seg[sg] >= K) {
                for (int b = sg * 32 + 31; b >= sg * 32; --b) {
                    acc += h[b];
                    if (acc >= K) { t = (uint32_t)b; found = true; break; }
                }
            } else {
                acc += seg[sg];
            }
        }
        thresh[row] = t;   // keep all elements with bin >= t  (count in [K, K+hist[t]))
    }
}

// ---------------- Kernel 4: compact surviving (key<<32 | ~idx) candidates -------
__global__ void k_compact(const float* __restrict__ scores,
                          const uint32_t* __restrict__ thresh,
                          uint32_t* __restrict__ cnt,
                          unsigned long long* __restrict__ cand) {
    const int row = blockIdx.x >> 6;                   // 64 blocks per row
    const int blk = blockIdx.x & 63;                   // 16384 elements per block
    const uint32_t t = thresh[row];
    const unsigned long long rowbase = (unsigned long long)row * N;
    const int tid = threadIdx.x;                       // 256 threads

    for (int it = 0; it < 16; ++it) {
        const uint32_t e = (uint32_t)blk * 16384u + (uint32_t)it * 1024u + (uint32_t)tid * 4u;
        const float* p = scores + rowbase + e;
        __builtin_prefetch((const void*)(p + 4096), 0, 0);   // global_prefetch_b8
        float4 v = *(const float4*)p;
        uint32_t ks[4] = { fkey(v.x), fkey(v.y), fkey(v.z), fkey(v.w) };
#pragma unroll
        for (int j = 0; j < 4; ++j) {
            if ((ks[j] >> BIN_SHIFT) >= t) {
                uint32_t pos = atomicAdd(&cnt[row], 1u);
                if (pos < CAND_CAP) {
                    cand[(size_t)row * CAND_CAP + pos] =
                        ((unsigned long long)ks[j] << 32) | (uint32_t)~(e + (uint32_t)j);
                }
            }
        }
    }
}

// ---------------- Kernel 5: LDS bitonic sort (desc) + emit top-K indices --------
__global__ __launch_bounds__(1024)
void k_sort(const uint32_t* __restrict__ cnt,
            const unsigned long long* __restrict__ cand,
            int* __restrict__ out) {
    __shared__ unsigned long long skey[SORT_N];        // 64 KB (WGP has 320 KB)
    const int row = blockIdx.x;
    const int tid = threadIdx.x;
    const int nth = blockDim.x;                        // 1024

    uint32_t M = cnt[row];
    if (M > CAND_CAP) M = CAND_CAP;
    const unsigned long long* c = cand + (size_t)row * CAND_CAP;
    for (int i = tid; i < SORT_N; i += nth)
        skey[i] = (i < (int)M) ? c[i] : 0ull;          // 0 sorts below all real keys
    __syncthreads();

    for (int size = 2; size <= SORT_N; size <<= 1) {
        for (int stride = size >> 1; stride > 0; stride >>= 1) {
            for (int i = tid; i < SORT_N; i += nth) {
                int j = i ^ stride;
                if (j > i) {
                    unsigned long long a = skey[i], b = skey[j];
                    bool dirDesc = ((i & size) == 0);
                    if ((a < b) == dirDesc) { skey[i] = b; skey[j] = a; }
                }
            }
            __syncthreads();
        }
    }

    for (int i = tid; i < K; i += nth)
        out[(size_t)row * K + i] = (int)~((uint32_t)skey[i]);
}

// ---------------- Host-side launch ----------------------------------------------
extern "C" void kernel_launch(void* const* d_in, const int* in_sizes, int n_in,
                              void* d_out, int out_size, void* d_ws, size_t ws_size,
                              hipStream_t stream) {
    (void)in_sizes; (void)n_in; (void)out_size; (void)ws_size;

    const float* scores = (const float*)d_in[0];
    int* out = (int*)d_out;

    char* ws = (char*)d_ws;
    uint32_t* hist             = (uint32_t*)ws;                         // 1 MB
    uint32_t* cnt              = (uint32_t*)(ws + HIST_BYTES);          // 128 B
    uint32_t* thresh           = (uint32_t*)(ws + HIST_BYTES + 256);    // 128 B
    unsigned long long* cand   = (unsigned long long*)(ws + HIST_BYTES + 4096); // 2 MB

    k_init   <<<(B * NBINS + 255) / 256, 256, 0, stream>>>(hist, cnt);
    k_hist   <<<B * 256, 256, 0, stream>>>(scores, hist);
    k_thresh <<<B,       256, 0, stream>>>(hist, thresh);
    k_compact<<<B * 64,  256, 0, stream>>>(scores, thresh, cnt, cand);
    k_sort   <<<B,      1024, 0, stream>>>(cnt, cand, out);
}